// PPRPowerIteration_8985071583608
// MI455X (gfx1250) — compile-verified
//
#include <hip/hip_runtime.h>
#include <stdint.h>

#define ALPHA 0.1f
#define NITER 10
#define D 64

// ---------------- CDNA5 async global->LDS staging ----------------
#if defined(__HIP_DEVICE_COMPILE__)
#if __has_builtin(__builtin_amdgcn_global_load_async_to_lds_b32)
#define USE_ASYNC_LDS 1
// Builtin signature (per hipcc diagnostic): (int* gsrc, int* ldst, imm offset, imm cpol)
#define ASYNC_B32(gptr, lptr) \
    __builtin_amdgcn_global_load_async_to_lds_b32((int*)(gptr), (int*)(lptr), 0, 0)
#else
#define USE_ASYNC_LDS 0
#endif
#if __has_builtin(__builtin_amdgcn_s_wait_asynccnt)
#define WAIT_ASYNC() do { __builtin_amdgcn_s_wait_asynccnt(0); asm volatile("" ::: "memory"); } while (0)
#else
#define WAIT_ASYNC() asm volatile("s_wait_asynccnt 0" ::: "memory")
#endif
#endif

// ---------------- Probe (NOT launched): placed first so the disasm snippet
// shows the exact async-to-LDS mnemonics this toolchain emits. ----------------
__global__ void k_async_probe(const int* __restrict__ g, int* __restrict__ out) {
    __shared__ int sh[32];
    const int lane = threadIdx.x & 31;
#if USE_ASYNC_LDS
    ASYNC_B32(g + lane, &sh[lane]);
    WAIT_ASYNC();
#else
    sh[lane] = g[lane];
#endif
    out[lane] = sh[lane];
}

// ---------------- SpMM: one row per wave32, lane owns 2 of 64 dims ----------------
// next[r] = alpha*E[r] + sum_e csr_val[e] * cur[csr_col[e]]
// Edge (col,val) chunks of 32 are staged into LDS with async global->LDS copies
// (ASYNCcnt-tracked), then consumed as LDS broadcasts while 256B-coalesced
// float2 row gathers stream from L2 (preds fits in the 192MB L2).
__global__ void __launch_bounds__(256)
k_spmm(const float* __restrict__ cur, const float* __restrict__ E,
       const int* __restrict__ row_start, const int* __restrict__ csr_col,
       const float* __restrict__ csr_val, float* __restrict__ next, int n) {
    __shared__ int   sh_col[8 * 32];
    __shared__ float sh_val[8 * 32];
    const int wave = threadIdx.x >> 5;
    const int lane = threadIdx.x & 31;
    const int r = blockIdx.x * 8 + wave;
    if (r >= n) return;  // wave-uniform

    const int s = row_start[r];
    const int e = row_start[r + 1];

    const float2 ev = *(const float2*)(E + (size_t)r * D + 2 * lane);
    float2 acc0 = make_float2(ALPHA * ev.x, ALPHA * ev.y);
    float2 acc1 = make_float2(0.0f, 0.0f);

    int*   mycol = sh_col + wave * 32;
    float* myval = sh_val + wave * 32;

    for (int base = s; base < e; base += 32) {
        const int k = base + lane;
#if USE_ASYNC_LDS
        if (k < e) {
            ASYNC_B32(csr_col + k, &mycol[lane]);
            ASYNC_B32(csr_val + k, &myval[lane]);
        }
        WAIT_ASYNC();
#else
        if (k < e) {
            mycol[lane] = csr_col[k];
            myval[lane] = csr_val[k];
        }
#endif
        const int cnt = min(32, e - base);
        int j = 0;
        for (; j + 1 < cnt; j += 2) {
            const int   c0 = mycol[j],     c1 = mycol[j + 1];
            const float v0 = myval[j],     v1 = myval[j + 1];
            const float2 p0 = *(const float2*)(cur + (size_t)c0 * D + 2 * lane);
            const float2 p1 = *(const float2*)(cur + (size_t)c1 * D + 2 * lane);
            acc0.x = fmaf(v0, p0.x, acc0.x);
            acc0.y = fmaf(v0, p0.y, acc0.y);
            acc1.x = fmaf(v1, p1.x, acc1.x);
            acc1.y = fmaf(v1, p1.y, acc1.y);
        }
        if (j < cnt) {
            const int   c0 = mycol[j];
            const float v0 = myval[j];
            const float2 p0 = *(const float2*)(cur + (size_t)c0 * D + 2 * lane);
            acc0.x = fmaf(v0, p0.x, acc0.x);
            acc0.y = fmaf(v0, p0.y, acc0.y);
        }
    }
    acc0.x += acc1.x;
    acc0.y += acc1.y;
    *(float2*)(next + (size_t)r * D + 2 * lane) = acc0;
}

// ---------------- CSR build: count ----------------
__global__ void k_count_rows(const int* __restrict__ edge_row, int nnz, int* __restrict__ cnt) {
    int i = blockIdx.x * blockDim.x + threadIdx.x;
    if (i < nnz) atomicAdd(&cnt[edge_row[i]], 1);
}

// ---------------- CSR build: single-block exclusive scan (N ~1e5, once per launch) ----------------
__global__ void k_scan(const int* __restrict__ cnt, int* __restrict__ row_start, int n, int nnz) {
    __shared__ int sh[1024];
    const int tid = threadIdx.x;
    int running = 0;
    for (int base = 0; base < n; base += 1024) {
        int i = base + tid;
        int v = (i < n) ? cnt[i] : 0;
        sh[tid] = v;
        __syncthreads();
        // Hillis-Steele inclusive scan
        for (int off = 1; off < 1024; off <<= 1) {
            int t = (tid >= off) ? sh[tid - off] : 0;
            __syncthreads();
            sh[tid] += t;
            __syncthreads();
        }
        int incl = sh[tid];
        if (i < n) row_start[i] = running + (incl - v);
        running += sh[1023];
        __syncthreads();  // protect sh before next chunk overwrites it
    }
    if (tid == 0) row_start[n] = nnz;
}

// ---------------- CSR build: scatter (fold 0.9 into values) ----------------
__global__ void k_scatter(const float* __restrict__ edge_val, const int* __restrict__ edge_row,
                          const int* __restrict__ edge_col, int nnz,
                          const int* __restrict__ row_start, int* __restrict__ fill,
                          int* __restrict__ csr_col, float* __restrict__ csr_val,
                          int* __restrict__ csr_idx) {
    int i = blockIdx.x * blockDim.x + threadIdx.x;
    if (i >= nnz) return;
    int r = edge_row[i];
    int pos = row_start[r] + atomicAdd(&fill[r], 1);
    csr_col[pos] = edge_col[i];
    csr_val[pos] = (1.0f - ALPHA) * edge_val[i];
    csr_idx[pos] = i;
}

// ---------------- Determinism: restore original edge order within each row ----------------
__global__ void k_sort_rows(const int* __restrict__ row_start, int* __restrict__ csr_col,
                            float* __restrict__ csr_val, int* __restrict__ csr_idx, int n) {
    int r = blockIdx.x * blockDim.x + threadIdx.x;
    if (r >= n) return;
    int s = row_start[r], e = row_start[r + 1];
    for (int i = s + 1; i < e; ++i) {
        int   ki = csr_idx[i];
        int   kc = csr_col[i];
        float kv = csr_val[i];
        int j = i - 1;
        while (j >= s && csr_idx[j] > ki) {
            csr_idx[j + 1] = csr_idx[j];
            csr_col[j + 1] = csr_col[j];
            csr_val[j + 1] = csr_val[j];
            --j;
        }
        csr_idx[j + 1] = ki;
        csr_col[j + 1] = kc;
        csr_val[j + 1] = kv;
    }
}

extern "C" void kernel_launch(void* const* d_in, const int* in_sizes, int n_in,
                              void* d_out, int out_size, void* d_ws, size_t ws_size,
                              hipStream_t stream) {
    const float* E        = (const float*)d_in[0];
    const float* edge_val = (const float*)d_in[1];
    const int*   edge_row = (const int*)d_in[2];
    const int*   edge_col = (const int*)d_in[3];

    const int ND  = in_sizes[0];
    const int N   = ND / D;
    const int NNZ = in_sizes[1];

    // Workspace layout (~45.6 MB for N=100k, NNZ=1.6M)
    char* p = (char*)d_ws;
    float* preds_b  = (float*)p; p += (size_t)ND * sizeof(float);
    float* csr_val  = (float*)p; p += (size_t)NNZ * sizeof(float);
    int*   csr_col  = (int*)p;   p += (size_t)NNZ * sizeof(int);
    int*   csr_idx  = (int*)p;   p += (size_t)NNZ * sizeof(int);
    int*   row_start= (int*)p;   p += (size_t)(N + 1) * sizeof(int);
    int*   cnt      = (int*)p;   p += (size_t)N * sizeof(int);  // reused as fill cursor

    const int TB = 256;
    const int gE = (NNZ + TB - 1) / TB;
    const int gN = (N + TB - 1) / TB;

    // --- Build deterministic CSR (once per launch; edges are iteration-invariant) ---
    (void)hipMemsetAsync(cnt, 0, (size_t)N * sizeof(int), stream);
    k_count_rows<<<gE, TB, 0, stream>>>(edge_row, NNZ, cnt);
    k_scan<<<1, 1024, 0, stream>>>(cnt, row_start, N, NNZ);
    (void)hipMemsetAsync(cnt, 0, (size_t)N * sizeof(int), stream);
    k_scatter<<<gE, TB, 0, stream>>>(edge_val, edge_row, edge_col, NNZ,
                                     row_start, cnt, csr_col, csr_val, csr_idx);
    k_sort_rows<<<gN, TB, 0, stream>>>(row_start, csr_col, csr_val, csr_idx, N);

    // --- Power iteration: ping-pong so iteration 9 (odd) lands in d_out ---
    float* buf0 = (float*)d_out;   // cur for even iterations
    float* buf1 = preds_b;
    (void)hipMemcpyAsync(buf0, E, (size_t)ND * sizeof(float), hipMemcpyDeviceToDevice, stream);

    const int gS = (N + 7) / 8;  // 8 rows (waves) per 256-thread block
    for (int it = 0; it < NITER; ++it) {
        const float* curp = (it & 1) ? buf1 : buf0;
        float*       nxtp = (it & 1) ? buf0 : buf1;
        k_spmm<<<gS, TB, 0, stream>>>(curp, E, row_start, csr_col, csr_val, nxtp, N);
    }
}